// AttnPool_18571438588282
// MI455X (gfx1250) — compile-verified
//
#include <hip/hip_runtime.h>

typedef __attribute__((ext_vector_type(2))) float v2f;
typedef __attribute__((ext_vector_type(8))) float v8f;

// ---- feature detection for the gfx1250 async global->LDS path ----
#ifndef USE_ASYNC_LDS
#if defined(__has_builtin)
#if __has_builtin(__builtin_amdgcn_global_load_async_to_lds_b128)
#define USE_ASYNC_LDS 1
#endif
#endif
#endif
#ifndef USE_ASYNC_LDS
#define USE_ASYNC_LDS 0
#endif

#define DMODEL 128
#define K1_ROWS 64     // rows staged per workgroup (4 waves x 16-row WMMA tiles)
#define K1_PAD  132    // 128 + 4 floats row padding -> conflict-free LDS column reads

#if USE_ASYNC_LDS
// Builtin expects (int4 addrspace(1)*, int4 addrspace(3)*, imm, imm); clang
// prints addrspace(1) as "__device__" in HIP diagnostics.
typedef int v4i_ __attribute__((vector_size(16)));
__device__ __forceinline__ void async_copy16(const float* gsrc, float* ldst) {
    __builtin_amdgcn_global_load_async_to_lds_b128(
        (__attribute__((address_space(1))) v4i_*)gsrc,
        (__attribute__((address_space(3))) v4i_*)ldst, 0, 0);
}
#endif

// -----------------------------------------------------------------------------
// Kernel 1: score[r] = dot(h[r,:], W) + b  via V_WMMA_F32_16X16X4_F32.
// Each wave owns 16 rows. A-fragment (16x4 f32): lanes 0-15 hold K=0,1 and
// lanes 16-31 hold K=2,3 of their row. B-fragment (4x16): column 0 carries the
// W chunk, other columns are zero, so D[:,0] accumulates the exact fp32 GEMV.
// -----------------------------------------------------------------------------
__global__ __launch_bounds__(128)
void attn_score_kernel(const float* __restrict__ h, const float* __restrict__ W,
                       const float* __restrict__ bptr, float* __restrict__ score,
                       int N) {
    __shared__ float tile[K1_ROWS * K1_PAD];
    __shared__ float sW[DMODEL];
    const int t = threadIdx.x;
    const long long rowBase = (long long)blockIdx.x * K1_ROWS;

    sW[t] = W[t];  // t in [0,128)

    // Stage 64x128 fp32 tile into LDS with coalesced 16B transfers.
#pragma unroll
    for (int it = 0; it < (K1_ROWS * (DMODEL / 4)) / 128; ++it) {
        int idx = it * 128 + t;
        int row = idx >> 5;      // 32 float4 per row
        int c4  = idx & 31;
        long long grow = rowBase + row;
        if (grow >= N) grow = N - 1;   // clamp: duplicate loads are harmless
        const float* src = h + grow * (long long)DMODEL + c4 * 4;
        float* dst = &tile[row * K1_PAD + c4 * 4];
#if USE_ASYNC_LDS
        async_copy16(src, dst);
#else
        *(float4*)dst = *(const float4*)src;
#endif
    }
#if USE_ASYNC_LDS
#if __has_builtin(__builtin_amdgcn_s_wait_asynccnt)
    __builtin_amdgcn_s_wait_asynccnt(0);
#else
    asm volatile("s_wait_asynccnt 0" ::: "memory");
#endif
#endif
    __syncthreads();

    const int wave = t >> 5;
    const int lane = t & 31;
    const int half = lane >> 4;    // selects K sub-pair {0,1} vs {2,3}
    const int mrow = lane & 15;    // M row within the 16-row tile
    const float* arow = &tile[(wave * 16 + mrow) * K1_PAD];
    const float wsel = (mrow == 0) ? 1.0f : 0.0f;  // only B column N=0 is live

    v8f acc = {};
#pragma unroll
    for (int k = 0; k < DMODEL; k += 4) {
        v2f a, bb;
        a.x  = arow[k + 2 * half];
        a.y  = arow[k + 2 * half + 1];
        bb.x = wsel * sW[k + 2 * half];
        bb.y = wsel * sW[k + 2 * half + 1];
        acc = __builtin_amdgcn_wmma_f32_16x16x4_f32(
            false, a, false, bb, (short)0, acc, false, false);
    }

    // D column N=0 lives in lane 0 (M = 0..7 across the 8 acc VGPRs) and
    // lane 16 (M = 8..15).
    if (mrow == 0) {
        const float bias = bptr[0];
        const long long r0 = rowBase + wave * 16 + half * 8;
#pragma unroll
        for (int j = 0; j < 8; ++j) {
            long long r = r0 + j;
            if (r < N) score[r] = acc[j] + bias;
        }
    }
}

// -----------------------------------------------------------------------------
// Kernel 2: one workgroup per segment (batch is sorted). Deterministic LDS
// tree reductions (no float atomics): max -> exp/sum -> weighted feature sum.
// -----------------------------------------------------------------------------
__device__ __forceinline__ int seg_lower_bound(const int* __restrict__ batch2,
                                               int n, int key) {
    int lo = 0, hi = n;
    while (lo < hi) {
        int mid = (lo + hi) >> 1;
        if (batch2[2 * mid] < key) lo = mid + 1; else hi = mid;  // int64 low word
    }
    return lo;
}

__global__ __launch_bounds__(128)
void attn_pool_kernel(const float* __restrict__ h, const int* __restrict__ batch2,
                      float* __restrict__ score, float* __restrict__ out, int N) {
    __shared__ float red[128];
    __shared__ int bounds[2];
    const int s = blockIdx.x;
    const int t = threadIdx.x;
    if (t == 0) bounds[0] = seg_lower_bound(batch2, N, s);
    if (t == 1) bounds[1] = seg_lower_bound(batch2, N, s + 1);
    __syncthreads();
    const int lo = bounds[0], hi = bounds[1];
    if (hi <= lo) {  // empty segment: reference emits a zero row
        out[(long long)s * DMODEL + t] = 0.0f;
        return;
    }

    // 1) segment max
    float lmax = -3.402823466e38f;
    for (int i = lo + t; i < hi; i += 128) lmax = fmaxf(lmax, score[i]);
    red[t] = lmax; __syncthreads();
#pragma unroll
    for (int off = 64; off > 0; off >>= 1) {
        if (t < off) red[t] = fmaxf(red[t], red[t + off]);
        __syncthreads();
    }
    const float m = red[0];
    __syncthreads();

    // 2) e = exp(score - m) (written back to ws), segment sum
    float lsum = 0.0f;
    for (int i = lo + t; i < hi; i += 128) {
        float e = __expf(score[i] - m);
        score[i] = e;
        lsum += e;
    }
    red[t] = lsum; __syncthreads();
#pragma unroll
    for (int off = 64; off > 0; off >>= 1) {
        if (t < off) red[t] += red[t + off];
        __syncthreads();
    }
    const float inv = 1.0f / red[0];

    // 3) weighted feature sum: one column per thread, coalesced row reads
    float acc = 0.0f;
    for (int i = lo; i < hi; ++i)
        acc += score[i] * h[(long long)i * DMODEL + t];
    out[(long long)s * DMODEL + t] = acc * inv;
}

// -----------------------------------------------------------------------------
extern "C" void kernel_launch(void* const* d_in, const int* in_sizes, int n_in,
                              void* d_out, int out_size, void* d_ws, size_t ws_size,
                              hipStream_t stream) {
    (void)n_in; (void)ws_size;
    const float* h      = (const float*)d_in[0];
    const int*   batch2 = (const int*)d_in[1];   // int64 viewed as LE int pairs
    const float* W      = (const float*)d_in[2];
    const float* bptr   = (const float*)d_in[3];
    float* out = (float*)d_out;
    const int N = in_sizes[1];
    const int B = out_size / DMODEL;
    float* score = (float*)d_ws;   // N floats of scratch

    attn_score_kernel<<<dim3((N + K1_ROWS - 1) / K1_ROWS), dim3(128), 0, stream>>>(
        h, W, bptr, score, N);
    attn_pool_kernel<<<dim3(B), dim3(128), 0, stream>>>(h, batch2, score, out, N);
}